// FPVNet_38328288150179
// MI455X (gfx1250) — compile-verified
//
#include <hip/hip_runtime.h>
#include <hip/hip_bf16.h>

// Problem constants (from the reference)
constexpr int Bc  = 8;
constexpr int Cc  = 5;
constexpr int Hc  = 256;
constexpr int Wc  = 704;
constexpr int HWc = Hc * Wc;            // 180224 (= 704 blocks of 256)
constexpr int NXc = 200, NYc = 200, NZc = 8;
constexpr int Vc  = NZc * NXc * NYc;    // 320000

#if __has_builtin(__builtin_amdgcn_global_load_async_to_lds_b32)
#define FPV_ASYNC_LDS 1
#else
#define FPV_ASYNC_LDS 0
#endif

// Builtin signature (from clang diagnostic): (AS1 int*, AS3 int*, imm, imm)
typedef __attribute__((address_space(1))) int as1_int;
typedef __attribute__((address_space(3))) int as3_int;

// ---------------------------------------------------------------------------
// Kernel 1: zero the (B,C,V) accumulator with wide B128 stores.
// ---------------------------------------------------------------------------
__global__ __launch_bounds__(256) void fpv_zero_f4(float4* __restrict__ out, int n4) {
    int i = blockIdx.x * 256 + threadIdx.x;
    if (i < n4) out[i] = make_float4(0.f, 0.f, 0.f, 0.f);
}

// ---------------------------------------------------------------------------
// Kernel 2: unproject + softmax + scatter-add.
// Grid: (HW/256, B). One pixel per thread; all lanes active (EXEC all-1s),
// which is required for the per-lane async global->LDS copies.
// ---------------------------------------------------------------------------
__global__ __launch_bounds__(256) void fpv_scatter(
    const float* __restrict__ depth,     // (B,1,H,W)
    const float* __restrict__ sem,       // (B,C,H,W)
    const float* __restrict__ K,         // (B,3,3)
    float*       __restrict__ out)       // (B,C,V)
{
    __shared__ float slog[Cc][256];

    const int tid = threadIdx.x;
    const int b   = blockIdx.y;
    const int p   = blockIdx.x * 256 + tid;       // pixel within batch

    // ---- Launch the 5 strided channel fetches first (async -> LDS) so they
    // overlap the depth load and all the index math below (ASYNCcnt path).
    const float* chan = sem + (size_t)b * Cc * HWc + p;
#if FPV_ASYNC_LDS
#pragma unroll
    for (int c = 0; c < Cc; ++c) {
        __builtin_amdgcn_global_load_async_to_lds_b32(
            (as1_int*)(chan + (size_t)c * HWc),
            (as3_int*)&slog[c][tid],
            /*imm offset*/0, /*cpol*/0);
    }
#else
#pragma unroll
    for (int c = 0; c < Cc; ++c) slog[c][tid] = chan[(size_t)c * HWc];
#endif

    // ---- Uniform intrinsics for this batch (blockIdx.y is uniform -> s_loads)
    const float fx  = K[b * 9 + 0];
    const float fy  = K[b * 9 + 4];
    const float cxk = K[b * 9 + 2];
    const float cyk = K[b * 9 + 5];

    // ---- Depth + unprojection (match reference rounding exactly: IEEE div,
    // trunc-toward-zero int casts, step 0.5 divisions == exact *2).
    float d = depth[(size_t)b * HWc + p];
    d = fminf(fmaxf(d, 1.0f), 50.0f);

    const int   vrow = p / Wc;
    const int   ucol = p - vrow * Wc;
    const float xc = ((float)ucol - cxk) * d / fx;
    const float yc = ((float)vrow - cyk) * d / fy;

    const int xi = (int)((xc + 50.0f) * 2.0f);   // (x - (-50)) / 0.5
    const int yi = (int)((d  + 50.0f) * 2.0f);   // (z - (-50)) / 0.5
    const int zi = (int)(yc + 2.0f);             // (y - (-2)) / 1.0

    const bool valid = (xi >= 0) & (xi < NXc) & (yi >= 0) & (yi < NYc)
                     & (zi >= 0) & (zi < NZc);
    const int flat = zi * (NXc * NYc) + xi * NYc + yi;

    // ---- Drain the async copies; each lane reads back only its own slots,
    // so a wave-level ASYNCcnt wait suffices (no workgroup barrier needed).
#if FPV_ASYNC_LDS
#if __has_builtin(__builtin_amdgcn_s_wait_asynccnt)
    __builtin_amdgcn_s_wait_asynccnt(0);
#else
    asm volatile("s_wait_asynccnt 0" ::: "memory");
#endif
    asm volatile("" ::: "memory");   // keep LDS reads below the wait
#endif

    float l[Cc];
#pragma unroll
    for (int c = 0; c < Cc; ++c) l[c] = slog[c][tid];

    // ---- Softmax over C=5 with max-subtraction (matches jax.nn.softmax)
    float m = l[0];
#pragma unroll
    for (int c = 1; c < Cc; ++c) m = fmaxf(m, l[c]);
    float e[Cc];
    float s = 0.0f;
#pragma unroll
    for (int c = 0; c < Cc; ++c) { e[c] = expf(l[c] - m); s += e[c]; }
    const float inv = 1.0f / s;

    // ---- Scatter: 5 native f32 global atomics (no return -> non-RTN form).
    // Output (51.2 MB) is L2-resident on MI455X (192 MB L2), so these
    // resolve in the L2 atomic units.
    if (valid) {
        float* ob = out + (size_t)b * Cc * Vc + flat;
#pragma unroll
        for (int c = 0; c < Cc; ++c) {
            atomicAdd(ob + (size_t)c * Vc, e[c] * inv);
        }
    }
}

// ---------------------------------------------------------------------------
extern "C" void kernel_launch(void* const* d_in, const int* in_sizes, int n_in,
                              void* d_out, int out_size, void* d_ws, size_t ws_size,
                              hipStream_t stream) {
    (void)in_sizes; (void)n_in; (void)d_ws; (void)ws_size;

    const float* depth = (const float*)d_in[0];   // (8,1,256,704)
    const float* sem   = (const float*)d_in[1];   // (8,5,256,704)
    const float* K     = (const float*)d_in[2];   // (8,3,3)
    float*       out   = (float*)d_out;           // (8,5,8,200,200) = 12.8M f32

    // Zero-init accumulator (out_size = 12,800,000, divisible by 4)
    const int n4 = out_size >> 2;
    fpv_zero_f4<<<(n4 + 255) / 256, 256, 0, stream>>>((float4*)out, n4);

    // Scatter: (HW/256 = 704) x (B = 8) blocks of 256 threads
    dim3 grid(HWc / 256, Bc);
    fpv_scatter<<<grid, 256, 0, stream>>>(depth, sem, K, out);
}